// Contrastive_30416958390420
// MI455X (gfx1250) — compile-verified
//
#include <hip/hip_runtime.h>
#include <hip/hip_bf16.h>

typedef __attribute__((ext_vector_type(2))) float v2f;
typedef __attribute__((ext_vector_type(8))) float v8f;

#define DDIM   512
#define TILE   16
#define STRIDE 516   // 512 + 4 pad floats -> conflict-free ds_load_b64 fragment reads

// One wave32 per block; each block handles a tile of 16 pairs.
__global__ __launch_bounds__(32)
void contrastive_tile_kernel(const float* __restrict__ emb,
                             const int*   __restrict__ pair_a,
                             const int*   __restrict__ pair_b,
                             const int*   __restrict__ pair_same,
                             float*       __restrict__ partial,
                             int nPairs)
{
    __shared__ __align__(16) float sDiff[TILE * STRIDE];  // ~32 KB
    __shared__ int sPA[TILE];
    __shared__ int sPB[TILE];

    const int lane = threadIdx.x;        // 0..31
    const int base = blockIdx.x * TILE;

    // Pair metadata (clamped so EXEC stays full for the WMMA loop).
    int pidx   = base + (lane & 15);
    int pclamp = pidx < nPairs ? pidx : (nPairs - 1);
    int mySame = pair_same[pclamp];
    if (lane < 16) {
        sPA[lane] = pair_a[pclamp];
        sPB[lane] = pair_b[pclamp];
    }
    __syncthreads();

    // ---- Phase A: gather rows with b128 loads, stage diff rows into LDS ----
    for (int p = 0; p < TILE; ++p) {
        const float4* ra = (const float4*)(emb + (size_t)sPA[p] * DDIM);
        const float4* rb = (const float4*)(emb + (size_t)sPB[p] * DDIM);
        float* dst = &sDiff[p * STRIDE];
        #pragma unroll
        for (int c = 0; c < 4; ++c) {
            int idx = lane + c * 32;               // 0..127 float4 chunks
            float4 av = ra[idx];
            float4 bv = rb[idx];
            float4 dv;
            dv.x = av.x - bv.x;
            dv.y = av.y - bv.y;
            dv.z = av.z - bv.z;
            dv.w = av.w - bv.w;
            *(float4*)&dst[idx * 4] = dv;
        }
    }
    __syncthreads();

    // ---- Phase B: Gram diagonal via V_WMMA_F32_16X16X4_F32 ----
    // A-fragment of Diff (16x4) and B-fragment of Diff^T (4x16) have identical
    // lane/VGPR layouts, so one v2f feeds both operands. D[p][p] = ||diff_p||^2.
    const int h   = lane >> 4;        // 0: K+{0,1}, 1: K+{2,3}
    const int row = lane & 15;
    const float* rp = &sDiff[row * STRIDE + 2 * h];

    v8f c0 = {}, c1 = {}, c2 = {}, c3 = {};   // 4 accumulators: break RAW chain
    for (int kk = 0; kk < DDIM / 4; kk += 4) {
        v2f f0 = *(const v2f*)(rp + (kk + 0) * 4);
        v2f f1 = *(const v2f*)(rp + (kk + 1) * 4);
        v2f f2 = *(const v2f*)(rp + (kk + 2) * 4);
        v2f f3 = *(const v2f*)(rp + (kk + 3) * 4);
        c0 = __builtin_amdgcn_wmma_f32_16x16x4_f32(false, f0, false, f0, (short)0, c0, false, false);
        c1 = __builtin_amdgcn_wmma_f32_16x16x4_f32(false, f1, false, f1, (short)0, c1, false, false);
        c2 = __builtin_amdgcn_wmma_f32_16x16x4_f32(false, f2, false, f2, (short)0, c2, false, false);
        c3 = __builtin_amdgcn_wmma_f32_16x16x4_f32(false, f3, false, f3, (short)0, c3, false, false);
    }
    v8f csum = (c0 + c1) + (c2 + c3);

    // Diagonal lives at: p<8 -> lane p, VGPR p ; p>=8 -> lane p+16, VGPR p-8.
    union { v8f v; float f[8]; } u; u.v = csum;
    int  myPair = (lane < 16) ? lane : (lane - 16);
    int  myR    = (lane < 16) ? lane : (lane - 24);
    bool valid  = ((lane < 8) || (lane >= 24)) && (base + myPair < nPairs);

    float loss = 0.f;
    if (valid) {
        float d2 = u.f[myR & 7];
        float d  = sqrtf(fmaxf(d2, 1e-12f));          // clamp(1e-12).sqrt()
        float s  = (float)mySame;
        float hinge = fmaxf(0.f, 1.0f - d);           // margin m = 1.0
        loss = (1.f - s) * hinge * hinge + s * d * d;
    }
    #pragma unroll
    for (int off = 16; off > 0; off >>= 1)
        loss += __shfl_xor(loss, off, 32);
    if (lane == 0) partial[blockIdx.x] = loss;
}

// Deterministic fixed-order final reduction (writes, never accumulates, so no
// init pass and no cross-call state).
__global__ __launch_bounds__(256)
void contrastive_reduce_kernel(const float* __restrict__ partial, int n,
                               float* __restrict__ out, float invP)
{
    __shared__ float s[256];
    float acc = 0.f;
    for (int i = threadIdx.x; i < n; i += 256) acc += partial[i];
    s[threadIdx.x] = acc;
    __syncthreads();
    for (int off = 128; off > 0; off >>= 1) {
        if (threadIdx.x < off) s[threadIdx.x] += s[threadIdx.x + off];
        __syncthreads();
    }
    if (threadIdx.x == 0) out[0] = s[0] * invP;
}

extern "C" void kernel_launch(void* const* d_in, const int* in_sizes, int n_in,
                              void* d_out, int out_size, void* d_ws, size_t ws_size,
                              hipStream_t stream)
{
    const float* emb  = (const float*)d_in[0];
    const int*   pa   = (const int*)d_in[1];
    const int*   pb   = (const int*)d_in[2];
    const int*   psam = (const int*)d_in[3];

    const int nPairs  = in_sizes[1];                 // P = 32768
    const int nBlocks = (nPairs + TILE - 1) / TILE;  // 2048 tiles
    float* partial = (float*)d_ws;                   // nBlocks * 4 bytes

    contrastive_tile_kernel<<<nBlocks, 32, 0, stream>>>(
        emb, pa, pb, psam, partial, nPairs);

    const float invP = (float)(1.0 / ((double)nPairs + 1e-10));
    contrastive_reduce_kernel<<<1, 256, 0, stream>>>(
        partial, nBlocks, (float*)d_out, invP);
}